// Transducer_40080634807166
// MI455X (gfx1250) — compile-verified
//
#include <hip/hip_runtime.h>
#include <stdint.h>

#define T_DIM 4000
#define B_DIM 64
#define C_DIM 64
#define U_DIM 200
#define NEG_INF (-1e30f)
#define RING    16      // LDS ring slots (each 64 floats = 256 B)
#define DEPTH   15      // async prefetch depth (prologue issues idx 0..14)

typedef __attribute__((ext_vector_type(2))) float v2f;
typedef __attribute__((ext_vector_type(8))) float v8f;

// logaddexp, mirrors jnp.logaddexp for finite inputs (incl. the -1e30 sentinel)
__device__ __forceinline__ float lae(float x, float y) {
    float m = fmaxf(x, y);
    float d = fabsf(x - y);
    return m + log1pf(__expf(-d));
}

// ---------------------------------------------------------------------------
// Kernel 1: lse[r] = logsumexp(inputs[r, 0:64]) for r = t*B + b  (256000 rows)
// One wave per row; wave32 shfl_xor reductions. Memory-bound (~65 MB total).
// ---------------------------------------------------------------------------
__global__ void lse_kernel(const float* __restrict__ in, float* __restrict__ lse) {
    int row  = blockIdx.x * 8 + (threadIdx.x >> 5);   // 8 waves / 256-thread block
    int lane = threadIdx.x & 31;
    const float* p = in + (size_t)row * C_DIM;
    float x0 = p[lane];
    float x1 = p[lane + 32];
    float m = fmaxf(x0, x1);
#pragma unroll
    for (int off = 16; off; off >>= 1) m = fmaxf(m, __shfl_xor(m, off, 32));
    float s = __expf(x0 - m) + __expf(x1 - m);
#pragma unroll
    for (int off = 16; off; off >>= 1) s += __shfl_xor(s, off, 32);
    if (lane == 0) lse[row] = m + __logf(s);
}

// ---------------------------------------------------------------------------
// Kernel 2: per-batch serial alpha scan. One wave per b. Lane l owns
// u = 7l .. 7l+6 (224 slots cover U=200; extras padded with NEG_INF-style
// finite sentinels, mirroring the reference arithmetic exactly).
// Logit rows stream into an LDS ring via gfx1250 async load-to-LDS, 14 steps
// ahead of consumption; s_wait_asynccnt gates slot readiness.
// ---------------------------------------------------------------------------
__global__ void scan_kernel(const float* __restrict__ in,
                            const int*  __restrict__ targets,
                            const float* __restrict__ lse,
                            float* __restrict__ losses) {
    __shared__ __align__(16) float s_lse[T_DIM];        // 16000 B
    __shared__ __align__(16) float ring[RING * C_DIM];  //  4096 B
    const int b    = blockIdx.x;
    const int lane = threadIdx.x;

    // Preload this batch's lse column (stride B) into LDS.
    for (int i = lane; i < T_DIM; i += 32)
        s_lse[i] = lse[(size_t)i * B_DIM + b];

    // Per-lane target class indices (loop-invariant gather addresses).
    int tgt[7];
#pragma unroll
    for (int i = 0; i < 7; ++i) {
        int u = lane * 7 + i;
        tgt[i] = (u < U_DIM) ? targets[b * U_DIM + u] : 0;
    }

    const uint32_t ring_base = (uint32_t)(uintptr_t)(&ring[0]);
    const uint64_t gbase     = (uint64_t)(uintptr_t)in;
    const int      bperm_idx = ((lane - 1) & 31) << 2;  // hoisted shuffle index

    // Async prologue: issue copies for rows 0..DEPTH-1. One b64 per lane
    // moves the whole 256 B row of inputs[t, b, :] into a ring slot.
#pragma unroll
    for (int idx = 0; idx < DEPTH; ++idx) {
        uint32_t va = (uint32_t)(((idx * B_DIM + b) * C_DIM) * 4 + lane * 8);
        uint32_t la = ring_base + (uint32_t)((((idx & (RING - 1)) * C_DIM) + lane * 2) * 4);
        asm volatile("global_load_async_to_lds_b64 %0, %1, %2"
                     :: "v"(la), "v"(va), "s"(gbase) : "memory");
    }
    __syncthreads();  // single wave: cheap; orders s_lse stores vs. reads

    // Wait until rows 0 and 1 have landed (issued 15, outstanding <= 13).
    asm volatile("s_wait_asynccnt 13" ::: "memory");

    // t = 0 peel: alpha[u]=NEG except alpha[0]=emit[0,b,0].
    float a[7];
#pragma unroll
    for (int i = 0; i < 7; ++i) a[i] = NEG_INF;
    {
        float e0 = ring[tgt[0]] - s_lse[0];
        if (lane == 0) a[0] = e0;
    }

    // Pre-gather row 1 into g (double-buffered gather).
    float g[7];
#pragma unroll
    for (int i = 0; i < 7; ++i) g[i] = ring[1 * C_DIM + tgt[i]];
    float glse = s_lse[1];

    for (int t = 1; t < T_DIM; ++t) {
        // Consume the pre-gathered row for this step.
        float e[7];
#pragma unroll
        for (int i = 0; i < 7; ++i) e[i] = g[i] - glse;

        // Start gathering row t+1 while the recurrence math runs below.
        // Issued = t+14 copies so far; need copy t+1 complete
        // => completed >= t+2 => outstanding <= 12.
        if (t + 1 < T_DIM) {
            asm volatile("s_wait_asynccnt 12" ::: "memory");
            int slot = (t + 1) & (RING - 1);
#pragma unroll
            for (int i = 0; i < 7; ++i) g[i] = ring[slot * C_DIM + tgt[i]];
            glse = s_lse[t + 1];
        }

        // alpha update: cross-lane shift via one ds_bpermute, then in-lane
        // descending so a[i-1] is still the previous step's value.
        float prev = __int_as_float(
            __builtin_amdgcn_ds_bpermute(bperm_idx, __float_as_int(a[6])));
        if (lane == 0) prev = NEG_INF;
#pragma unroll
        for (int i = 6; i >= 1; --i) a[i] = e[i] + lae(a[i], a[i - 1]);
        a[0] = e[0] + lae(a[0], prev);

        // Refill the ring DEPTH-1 steps ahead; its slot was consumed at t-2.
        int nidx = t + 14;
        if (nidx < T_DIM) {
            uint32_t va = (uint32_t)(((nidx * B_DIM + b) * C_DIM) * 4 + lane * 8);
            uint32_t la = ring_base + (uint32_t)((((nidx & (RING - 1)) * C_DIM) + lane * 2) * 4);
            asm volatile("global_load_async_to_lds_b64 %0, %1, %2"
                         :: "v"(la), "v"(va), "s"(gbase) : "memory");
        }
    }

    // u = U-1 = 199 lives at lane 28, slot 3.
    if (lane == 28) losses[b] = -a[3];
}

// ---------------------------------------------------------------------------
// Kernel 3: mean of the 64 per-batch losses via V_WMMA_F32_16X16X4_F32.
// A = losses as a 16x4 matrix (ISA A-striping: lanes 0-15 hold K=0,1 in
// VGPR0,1; lanes 16-31 hold K=2,3). B = ones(4x16), so D[m][n] = row-sum(m)
// independent of B's striping. Rows 0-7 -> lanes 0-15, rows 8-15 -> lanes
// 16-31 across the 8 accumulator components; per-lane component sum + one
// shfl_xor(16) yields the deterministic total. One wave, EXEC all ones.
// ---------------------------------------------------------------------------
__global__ void finalize_kernel(const float* __restrict__ losses, float* __restrict__ out) {
    int lane = threadIdx.x;
    int base = (lane & 15) * 4 + (lane >> 4) * 2;   // row = lane&15, K pair by half
    v2f A;
    A.x = losses[base];
    A.y = losses[base + 1];
    v2f Bm;
    Bm.x = 1.0f;
    Bm.y = 1.0f;
    v8f C = {};
    C = __builtin_amdgcn_wmma_f32_16x16x4_f32(
        /*neg_a=*/false, A, /*neg_b=*/false, Bm,
        /*c_mod=*/(short)0, C, /*reuse_a=*/false, /*reuse_b=*/false);
    float s = C[0] + C[1] + C[2] + C[3] + C[4] + C[5] + C[6] + C[7];
    s += __shfl_xor(s, 16, 32);
    if (lane == 0) out[0] = s * (1.0f / (float)B_DIM);
}

extern "C" void kernel_launch(void* const* d_in, const int* in_sizes, int n_in,
                              void* d_out, int out_size, void* d_ws, size_t ws_size,
                              hipStream_t stream) {
    (void)in_sizes; (void)n_in; (void)out_size; (void)ws_size;
    const float* inputs  = (const float*)d_in[0];   // (T, B, C) f32
    const int*   targets = (const int*)d_in[1];     // (B, U)    i32

    float* lse    = (float*)d_ws;                                       // T*B floats = 1 MB
    float* losses = (float*)((char*)d_ws + (size_t)T_DIM * B_DIM * sizeof(float)); // 64 floats

    // 256000 rows, 8 waves (one row each) per 256-thread block.
    lse_kernel<<<(T_DIM * B_DIM) / 8, 256, 0, stream>>>(inputs, lse);

    // One wave32 per batch element.
    scan_kernel<<<B_DIM, 32, 0, stream>>>(inputs, targets, lse, losses);

    finalize_kernel<<<1, 32, 0, stream>>>(losses, (float*)d_out);
}